// FlowMSELoss_60000693125250
// MI455X (gfx1250) — compile-verified
//
#include <hip/hip_runtime.h>
#include <hip/hip_bf16.h>

// Problem constants (fixed by the reference: B=8, C=3, T=16, H=W=224, P0=2, P=16, FC=2)
#define BB   8
#define CC   3
#define TT   16
#define HH   224
#define WW   224
#define HW   (HH * WW)          // 50176
#define NTOK 1568               // (T/2)*(H/16)*(W/16) = 8*14*14
#define DTOK 1024               // 2*16*16*2
#define NPOS (BB * TT * HH * WW) // 6,422,528 spatial positions (c handled in-register)
#define NPART 2048              // block partials, multiple of 64 for WMMA reduction
#define THREADS 256

typedef __attribute__((ext_vector_type(2))) float v2f;
typedef __attribute__((ext_vector_type(8))) float v8f;

// ---------------------------------------------------------------------------
// ws layout (floats): ws[0] = mask count, ws[1 .. NPART] = block partial sums
// ---------------------------------------------------------------------------

__global__ void fml_zero_ws(float* __restrict__ ws, int n) {
    int i = blockIdx.x * blockDim.x + threadIdx.x;
    if (i < n) ws[i] = 0.0f;
}

__device__ __forceinline__ float wave32_sum(float v) {
    #pragma unroll
    for (int off = 16; off > 0; off >>= 1) v += __shfl_down(v, off, 32);
    return v;
}

// ---- mask popcount: B*NTOK = 12544 ints -> ws[0] --------------------------
__global__ void fml_mask_count(const int* __restrict__ mask, float* __restrict__ ws) {
    float local = 0.0f;
    for (int i = blockIdx.x * blockDim.x + threadIdx.x; i < BB * NTOK;
         i += gridDim.x * blockDim.x)
        local += (mask[i] != 0) ? 1.0f : 0.0f;

    local = wave32_sum(local);
    __shared__ float sm[8];
    int lane = threadIdx.x & 31, wv = threadIdx.x >> 5;
    if (lane == 0) sm[wv] = local;
    __syncthreads();
    if (threadIdx.x < 32) {
        float v = (threadIdx.x < (THREADS >> 5)) ? sm[threadIdx.x] : 0.0f;
        #pragma unroll
        for (int off = 4; off > 0; off >>= 1) v += __shfl_down(v, off, 32);
        if (threadIdx.x == 0) atomicAdd(ws, v);
    }
}

// ---- fused bilinear-warp + masked squared-diff partial reduction ----------
__global__ void fml_warp_mse(const float* __restrict__ tok,   // (B, NTOK, DTOK)
                             const float* __restrict__ img,   // (B, C, T, H, W)
                             const int*   __restrict__ mask,  // (B, NTOK)
                             float* __restrict__ ws) {
    float acc = 0.0f;
    const int stride = gridDim.x * blockDim.x;
    for (int s = blockIdx.x * blockDim.x + threadIdx.x; s < NPOS; s += stride) {
        int w = s % WW;
        int r = s / WW;
        int h = r % HH;  r /= HH;
        int t = r % TT;
        int b = r / TT;

        int n = (t >> 1) * 196 + (h >> 4) * 14 + (w >> 4);
        if (mask[b * NTOK + n] == 0) continue;  // masked-out token: zero contribution

        // flow read (unpatchify is pure indexing)
        int fi = (b * NTOK + n) * DTOK + ((((t & 1) << 4) + (h & 15)) << 5) + ((w & 15) << 1);
        float fx = tok[fi];
        float fy = tok[fi + 1];

        float mx = fx + (float)w;
        float my = fy + (float)h;
        float x0f = floorf(mx), y0f = floorf(my);
        float wx = mx - x0f, wy = my - y0f;
        int x0 = (int)x0f, y0 = (int)y0f;
        int x1 = x0 + 1,   y1 = y0 + 1;

        bool vx0 = (x0 >= 0) & (x0 < WW);
        bool vx1 = (x1 >= 0) & (x1 < WW);
        bool vy0 = (y0 >= 0) & (y0 < HH);
        bool vy1 = (y1 >= 0) & (y1 < HH);
        int xc0 = min(max(x0, 0), WW - 1), xc1 = min(max(x1, 0), WW - 1);
        int yc0 = min(max(y0, 0), HH - 1), yc1 = min(max(y1, 0), HH - 1);

        float w00 = (1.0f - wx) * (1.0f - wy);
        float w01 = wx * (1.0f - wy);
        float w10 = (1.0f - wx) * wy;
        float w11 = wx * wy;

        int i00 = yc0 * WW + xc0, i01 = yc0 * WW + xc1;
        int i10 = yc1 * WW + xc0, i11 = yc1 * WW + xc1;
        int ic  = h * WW + w;
        int base = ((b * CC) * TT + t) * HW;  // channel 0 plane of frame t

        #pragma unroll
        for (int c = 0; c < CC; ++c) {
            const float* p = img + base + c * (TT * HW);
            float v00 = (vy0 & vx0) ? p[i00] : 0.0f;
            float v01 = (vy0 & vx1) ? p[i01] : 0.0f;
            float v10 = (vy1 & vx0) ? p[i10] : 0.0f;
            float v11 = (vy1 & vx1) ? p[i11] : 0.0f;
            float rec = v00 * w00 + v01 * w01 + v10 * w10 + v11 * w11;
            float d = rec - p[ic];
            acc = fmaf(d, d, acc);
        }
    }

    // block reduction: wave32 shuffles + LDS across 8 waves
    acc = wave32_sum(acc);
    __shared__ float sm[8];
    int lane = threadIdx.x & 31, wv = threadIdx.x >> 5;
    if (lane == 0) sm[wv] = acc;
    __syncthreads();
    if (threadIdx.x < 32) {
        float v = (threadIdx.x < (THREADS >> 5)) ? sm[threadIdx.x] : 0.0f;
        #pragma unroll
        for (int off = 4; off > 0; off >>= 1) v += __shfl_down(v, off, 32);
        if (threadIdx.x == 0) ws[1 + blockIdx.x] = v;
    }
}

// ---- finalize: WMMA f32 tree-reduction of NPART partials, one wave32 ------
// A = all-ones 16x4 f32 -> D[m][n] = sum_k B[k][n]; accumulating 4x16 B tiles
// (64 partials each) into C gives exact-f32 column sums. Every loaded value
// occupies a distinct (k,n) cell of B, so the grand total is Sum_n D[0][n].
// Each column sum appears in rows 0 and 8 of C's VGPR0, so summing c[0] over
// all 32 lanes yields 2x the total. EXEC is all-1s (32 convergent lanes).
__global__ void fml_finalize(const float* __restrict__ ws, float* __restrict__ out) {
    int lane = threadIdx.x;           // 0..31, single wave32
    v2f a; a.x = 1.0f; a.y = 1.0f;    // ones A-matrix (16x4, 2 VGPRs)
    v8f c = {};
    const float* parts = ws + 1;
    #pragma unroll
    for (int it = 0; it < NPART / 64; ++it) {
        v2f bm;
        bm.x = parts[it * 64 + lane];
        bm.y = parts[it * 64 + 32 + lane];
        // v_wmma_f32_16x16x4_f32: exact f32 FMA, RNE — no precision loss
        c = __builtin_amdgcn_wmma_f32_16x16x4_f32(
                /*neg_a=*/false, a, /*neg_b=*/false, bm,
                /*c_mod=*/(short)0, c, /*reuse_a=*/false, /*reuse_b=*/false);
    }
    float s = c[0];
    #pragma unroll
    for (int off = 16; off > 0; off >>= 1) s += __shfl_down(s, off, 32);
    if (lane == 0) {
        float total = s * 0.5f;                 // rows 0 and 8 both counted
        float denom = fmaxf(ws[0], 1.0f) * 1536.0f;  // max(mask.sum(),1) * p0*p1*p2*C
        out[0] = total / denom;
    }
}

extern "C" void kernel_launch(void* const* d_in, const int* in_sizes, int n_in,
                              void* d_out, int out_size, void* d_ws, size_t ws_size,
                              hipStream_t stream) {
    const float* tok  = (const float*)d_in[0];  // output tokens (B, N, D) f32
    const float* img  = (const float*)d_in[1];  // raw_target (B, C, T, H, W) f32
    const int*   mask = (const int*)d_in[2];    // (B, N) int
    float* out = (float*)d_out;
    float* ws  = (float*)d_ws;

    (void)in_sizes; (void)n_in; (void)out_size; (void)ws_size;

    fml_zero_ws<<<(NPART + 1 + THREADS - 1) / THREADS, THREADS, 0, stream>>>(ws, NPART + 1);
    fml_mask_count<<<49, THREADS, 0, stream>>>(mask, ws);
    fml_warp_mse<<<NPART, THREADS, 0, stream>>>(tok, img, mask, ws);
    fml_finalize<<<1, 32, 0, stream>>>(ws, out);
}